// LearnableSineOscillator_78709570666739
// MI455X (gfx1250) — compile-verified
//
#include <hip/hip_runtime.h>
#include <math.h>

// CDNA5 (gfx1250) wave32 WMMA types
typedef __attribute__((ext_vector_type(2))) float v2f;
typedef __attribute__((ext_vector_type(4))) float f4;
typedef __attribute__((ext_vector_type(8))) float v8f;

__device__ __forceinline__ void cmul_inplace(float& ar, float& ai, float br, float bi) {
    float r = fmaf(ar, br, -ai * bi);
    float i = fmaf(ar, bi,  ai * br);
    ar = r; ai = i;
}

// (rr,ri) = (br,bi)^e by binary exponentiation
__device__ __forceinline__ void cpowu(float br, float bi, unsigned e, float& rr, float& ri) {
    rr = 1.0f; ri = 0.0f;
    while (e) {
        if (e & 1u) cmul_inplace(rr, ri, br, bi);
        e >>= 1;
        if (e) {
            float nr = fmaf(br, br, -bi * bi);
            float ni = 2.0f * br * bi;
            br = nr; bi = ni;
        }
    }
}

// constrain: zc = z * tanh(|z|)/|z|
__device__ __forceinline__ void constrain_z(float& zr, float& zi) {
    float mag = sqrtf(fmaf(zr, zr, zi * zi));
    float sc  = tanhf(mag) / mag;
    zr *= sc; zi *= sc;
}

// y[k] = Re(phase * zc^k).
// Each wave32 produces 256-output tiles; one V_WMMA_F32_16X16X4_F32 per tile:
//   D[m][n] = s_m.re*p_n.re - s_m.im*p_n.im = Re(s_m * p_n)
//   s_m = phase*zc^(base+m)   (A: lanes 0-15 hold K=0..1, lanes 16-31 zero pad)
//   p_n = (zc^16)^n           (B: lanes 0-15,             lanes 16-31 zero pad)
// Output index = base + m + 16*n -> lane's 8 D VGPRs are 8 contiguous floats
// -> two non-temporal global_store_b128 per lane per tile (fully coalesced).
// Trip count is broadcast via readfirstlane so the loop is SGPR-controlled and
// EXEC stays all-1s at every WMMA by construction.
__global__ __launch_bounds__(256) void
LearnableSineOscillator_wmma_kernel(const float* __restrict__ zrp,
                                    const float* __restrict__ zip,
                                    const float* __restrict__ prp,
                                    const float* __restrict__ pip,
                                    float* __restrict__ out,
                                    unsigned fullTiles, unsigned tilesPerWave)
{
    float zr = zrp[0], zi = zip[0];
    float pr = prp[0], pi = pip[0];
    constrain_z(zr, zi);

    unsigned tid  = blockIdx.x * blockDim.x + threadIdx.x;
    unsigned wave = tid >> 5;               // wave32
    unsigned lane = tid & 31u;
    unsigned m    = lane & 15u;
    bool     loH  = (lane < 16u);

    // Wave-uniform tile range; force into SGPR so the loop never touches EXEC.
    unsigned waveTile0 = wave * tilesPerWave;
    unsigned myTilesV  = (waveTile0 < fullTiles)
                       ? ((fullTiles - waveTile0 < tilesPerWave) ? (fullTiles - waveTile0)
                                                                 : tilesPerWave)
                       : 0u;
    unsigned myTiles = (unsigned)__builtin_amdgcn_readfirstlane((int)myTilesV);

    // zc^16 and zc^256 (tile-to-tile step)
    float w16r, w16i;   cpowu(zr, zi, 16u, w16r, w16i);
    float w256r, w256i; cpowu(w16r, w16i, 16u, w256r, w256i);

    // B matrix (constant per wave): lanes 0-15 = { (zc^16)^n .re, .im }, else 0
    float bnr = 0.0f, bni = 0.0f;
    if (loH) cpowu(w16r, w16i, m, bnr, bni);
    v2f B; B.x = bnr; B.y = bni;

    // Seed s = phase * zc^(waveBase + m); zeroed in lanes 16-31 so the A
    // operand needs no per-iteration masking (s *= zc^256 preserves zeros).
    unsigned base = waveTile0 * 256u;
    float tr, ti; cpowu(zr, zi, base + m, tr, ti);
    float sr = loH ? fmaf(pr, tr, -pi * ti) : 0.0f;
    float si = loH ? fmaf(pr, ti,  pi * tr) : 0.0f;

    // Lane's store offset within its tile: 8*(lane>=16) + 16*(lane&15)
    unsigned laneOff = ((lane & 16u) >> 1) + (m << 4);
    float* p = out + base + laneOff;                 // 16B aligned

    for (unsigned t = 0; t < myTiles; ++t, p += 256) {
        // A matrix: { s_m.re, -s_m.im } (lanes 16-31 carry zeros -> K=2,3 = 0)
        v2f A; A.x = sr; A.y = -si;

        v8f c = {0.f, 0.f, 0.f, 0.f, 0.f, 0.f, 0.f, 0.f};
        // 8 args: (neg_a, A, neg_b, B, c_mod, C, reuse_a, reuse_b)
        v8f d = __builtin_amdgcn_wmma_f32_16x16x4_f32(
            false, A, false, B, (short)0, c, false, false);

        f4 q0 = {d[0], d[1], d[2], d[3]};
        f4 q1 = {d[4], d[5], d[6], d[7]};
        __builtin_nontemporal_store(q0, reinterpret_cast<f4*>(p));
        __builtin_nontemporal_store(q1, reinterpret_cast<f4*>(p) + 1);

        // advance seed: s *= zc^256
        float nsr = fmaf(sr, w256r, -si * w256i);
        float nsi = fmaf(sr, w256i,  si * w256r);
        sr = nsr; si = nsi;
    }
}

// Scalar tail for n % 256 elements (not taken for N = 2^25; kept for generality).
__global__ __launch_bounds__(256) void
LearnableSineOscillator_tail_kernel(const float* __restrict__ zrp,
                                    const float* __restrict__ zip,
                                    const float* __restrict__ prp,
                                    const float* __restrict__ pip,
                                    float* __restrict__ out,
                                    unsigned start, unsigned n)
{
    unsigned k = start + blockIdx.x * blockDim.x + threadIdx.x;
    if (k >= n) return;
    float zr = zrp[0], zi = zip[0];
    float pr = prp[0], pi = pip[0];
    constrain_z(zr, zi);
    float tr, ti; cpowu(zr, zi, k, tr, ti);
    out[k] = fmaf(pr, tr, -pi * ti);
}

extern "C" void kernel_launch(void* const* d_in, const int* in_sizes, int n_in,
                              void* d_out, int out_size, void* d_ws, size_t ws_size,
                              hipStream_t stream) {
    (void)in_sizes; (void)n_in; (void)d_ws; (void)ws_size;
    const float* zrp = (const float*)d_in[0];
    const float* zip = (const float*)d_in[1];
    const float* prp = (const float*)d_in[2];
    const float* pip = (const float*)d_in[3];
    float* out = (float*)d_out;

    unsigned n         = (unsigned)out_size;     // N = 2^25
    unsigned fullTiles = n / 256u;               // 131072
    unsigned rem       = n % 256u;               // 0 for this reference

    const int threads = 256;                     // 8 wave32s / block
    const int blocks  = 1024;                    // 8192 waves total
    unsigned wavesTotal   = (unsigned)(blocks * threads) / 32u;
    unsigned tilesPerWave = (fullTiles + wavesTotal - 1u) / wavesTotal;  // 16
    if (tilesPerWave == 0u) tilesPerWave = 1u;

    if (fullTiles) {
        LearnableSineOscillator_wmma_kernel<<<blocks, threads, 0, stream>>>(
            zrp, zip, prp, pip, out, fullTiles, tilesPerWave);
    }
    if (rem) {
        LearnableSineOscillator_tail_kernel<<<1, 256, 0, stream>>>(
            zrp, zip, prp, pip, out, fullTiles * 256u, n);
    }
}